// RigidLoss_87170656240057
// MI455X (gfx1250) — compile-verified
//
#include <hip/hip_runtime.h>
#include <hip/hip_bf16.h>
#include <stdint.h>

// Problem constants (match reference)
#define N_PTS 4194304
#define K_CL  256
#define M_PTS 8192
#define TILE  1024          // points staged in LDS per block iteration
#define LPAD  (TILE + 4)    // padded LDS row stride (+4 floats -> +4 banks/row)
#define PART_A 1024
#define PART_B 1024

typedef __attribute__((ext_vector_type(2))) float v2f;
typedef __attribute__((ext_vector_type(4))) float v4f;
typedef __attribute__((ext_vector_type(8))) float v8f;

// Per-lane async copy: 4 bytes global -> LDS, tracked by ASYNCcnt.
__device__ __forceinline__ void async_g2l_b32(uint32_t lds_off, const float* gaddr) {
  asm volatile("global_load_async_to_lds_b32 %0, %1, off"
               :: "v"(lds_off), "v"((uint64_t)(uintptr_t)gaddr)
               : "memory");
}

// ---------------------------------------------------------------------------
// Phase 1: per-cluster fused reductions via V_WMMA_F32_16X16X4_F32.
// A (16x4): rows 0-2 = src x/y/z, row 3 = 1, rows 4-15 = 0. 4 points = K dim.
// B (4x16): cols 0-2 = tgt x/y/z (tgt = src + delta), col 3 = 1.
// Accumulator D[0..3][0..3] = {Σ s_i t_j, Σ s_i, Σ t_j, count}, full fp32.
// One block (8 wave32) per cluster; each wave owns a disjoint 1/8 of the tile.
// Gather is staged with GLOBAL_LOAD_ASYNC_TO_LDS_B32 (per-lane addresses).
// ---------------------------------------------------------------------------
__global__ __launch_bounds__(256) void rigid_phase1_sums(
    const float* __restrict__ pc_t, const float* __restrict__ sf_t,
    const int* __restrict__ cidx, float* __restrict__ sums) {
  __shared__ float ls[3][LPAD];   // src coords (pc_t gather)
  __shared__ float ld[3][LPAD];   // delta coords (sf_t gather)
  __shared__ float red[8][16];    // per-wave 4x4 accumulator blocks

  const int cluster = blockIdx.x;
  const int tid  = threadIdx.x;
  const int lane = tid & 31;
  const int wave = tid >> 5;

  // wave32 WMMA operand mapping (branchless)
  const int   r    = lane & 15;              // A row / B col this lane feeds
  const int   kg   = (lane >> 4) * 2;        // K-slot base: 0 or 2
  const int   rc   = (r < 3) ? r : 0;        // clamped LDS row
  const float msk  = (r < 3) ? 1.0f : 0.0f;  // coordinate rows
  const float onef = (r == 3) ? 1.0f : 0.0f; // homogeneous row/col
  const float* srow = &ls[rc][0];
  const float* drow = &ld[rc][0];

  v8f acc0 = {};
  v8f acc1 = {};

  for (int tile = 0; tile < M_PTS; tile += TILE) {
    // Async gather of the tile into LDS (6 streams, per-lane addresses).
    for (int t = tid; t < TILE; t += 256) {
      const int idx = cidx[cluster * M_PTS + tile + t];
      async_g2l_b32((uint32_t)(uintptr_t)&ls[0][t], pc_t + idx);
      async_g2l_b32((uint32_t)(uintptr_t)&ls[1][t], pc_t + N_PTS + idx);
      async_g2l_b32((uint32_t)(uintptr_t)&ls[2][t], pc_t + 2 * N_PTS + idx);
      async_g2l_b32((uint32_t)(uintptr_t)&ld[0][t], sf_t + idx);
      async_g2l_b32((uint32_t)(uintptr_t)&ld[1][t], sf_t + N_PTS + idx);
      async_g2l_b32((uint32_t)(uintptr_t)&ld[2][t], sf_t + 2 * N_PTS + idx);
    }
    asm volatile("s_wait_asynccnt 0x0" ::: "memory");
    __syncthreads();

    // Each wave consumes TILE/8 = 128 points, 4 per WMMA (32 WMMAs/tile),
    // alternating two accumulators to break the C-chain.
    const int base = wave * (TILE / 8);
#pragma unroll
    for (int c = 0; c < TILE / 8; c += 8) {
      {
        const int p = base + c + kg;
        const v2f s2 = *(const v2f*)(srow + p);   // ds_load_b64
        const v2f d2 = *(const v2f*)(drow + p);   // ds_load_b64
        v2f a, b;
        a.x = fmaf(s2.x, msk, onef);
        a.y = fmaf(s2.y, msk, onef);
        b.x = fmaf(s2.x + d2.x, msk, onef);
        b.y = fmaf(s2.y + d2.y, msk, onef);
        acc0 = __builtin_amdgcn_wmma_f32_16x16x4_f32(
            false, a, false, b, (short)0, acc0, false, false);
      }
      {
        const int p = base + c + 4 + kg;
        const v2f s2 = *(const v2f*)(srow + p);
        const v2f d2 = *(const v2f*)(drow + p);
        v2f a, b;
        a.x = fmaf(s2.x, msk, onef);
        a.y = fmaf(s2.y, msk, onef);
        b.x = fmaf(s2.x + d2.x, msk, onef);
        b.y = fmaf(s2.y + d2.y, msk, onef);
        acc1 = __builtin_amdgcn_wmma_f32_16x16x4_f32(
            false, a, false, b, (short)0, acc1, false, false);
      }
    }
    __syncthreads();
  }

  const v8f acc = acc0 + acc1;

  // Extract D[0..3][0..3]: VGPR i at lanes 0-15 holds row M=i, col N=lane.
  if (lane < 4) {
    red[wave][0 * 4 + lane] = acc[0];
    red[wave][1 * 4 + lane] = acc[1];
    red[wave][2 * 4 + lane] = acc[2];
    red[wave][3 * 4 + lane] = acc[3];
  }
  __syncthreads();
  if (tid < 16) {
    float s = 0.0f;
    for (int w = 0; w < 8; ++w) s += red[w][tid];
    sums[cluster * 16 + tid] = s;  // D[i][j] at i*4+j
  }
}

// ---------------------------------------------------------------------------
// Phase 2: per-cluster Kabsch via Horn quaternion (max eigvec of N(H) by
// shifted power iteration). 256 clusters -> 1 block of 256 threads.
// ---------------------------------------------------------------------------
__global__ __launch_bounds__(256) void rigid_phase2_kabsch(
    const float* __restrict__ sums, float* __restrict__ RT) {
  const int k = blockIdx.x * blockDim.x + threadIdx.x;
  if (k >= K_CL) return;
  const float* S = &sums[k * 16];
  const float invM = 1.0f / (float)M_PTS;
  const float m1[3] = { S[3] * invM, S[7] * invM, S[11] * invM };
  const float m2[3] = { S[12] * invM, S[13] * invM, S[14] * invM };
  float H[3][3];
  float fro = 0.0f;
  for (int i = 0; i < 3; ++i)
    for (int j = 0; j < 3; ++j) {
      H[i][j] = S[i * 4 + j] - (float)M_PTS * m1[i] * m2[j];
      fro += H[i][j] * H[i][j];
    }
  const float n00 =  H[0][0] + H[1][1] + H[2][2];
  const float n01 =  H[1][2] - H[2][1];
  const float n02 =  H[2][0] - H[0][2];
  const float n03 =  H[0][1] - H[1][0];
  const float n11 =  H[0][0] - H[1][1] - H[2][2];
  const float n12 =  H[0][1] + H[1][0];
  const float n13 =  H[0][2] + H[2][0];
  const float n22 = -H[0][0] + H[1][1] - H[2][2];
  const float n23 =  H[1][2] + H[2][1];
  const float n33 = -H[0][0] - H[1][1] + H[2][2];
  const float sig = 2.0f * sqrtf(fro) + 1e-12f;

  float q0 = 1.0f, q1 = 0.0f, q2 = 0.0f, q3 = 0.0f;
  for (int it = 0; it < 64; ++it) {
    const float r0 = (n00 + sig) * q0 + n01 * q1 + n02 * q2 + n03 * q3;
    const float r1 = n01 * q0 + (n11 + sig) * q1 + n12 * q2 + n13 * q3;
    const float r2 = n02 * q0 + n12 * q1 + (n22 + sig) * q2 + n23 * q3;
    const float r3 = n03 * q0 + n13 * q1 + n23 * q2 + (n33 + sig) * q3;
    const float rn = rsqrtf(r0 * r0 + r1 * r1 + r2 * r2 + r3 * r3 + 1e-30f);
    q0 = r0 * rn; q1 = r1 * rn; q2 = r2 * rn; q3 = r3 * rn;
  }
  const float w = q0, x = q1, y = q2, z = q3;
  float R[3][3];
  R[0][0] = 1.0f - 2.0f * (y * y + z * z);
  R[0][1] = 2.0f * (x * y - w * z);
  R[0][2] = 2.0f * (x * z + w * y);
  R[1][0] = 2.0f * (x * y + w * z);
  R[1][1] = 1.0f - 2.0f * (x * x + z * z);
  R[1][2] = 2.0f * (y * z - w * x);
  R[2][0] = 2.0f * (x * z - w * y);
  R[2][1] = 2.0f * (y * z + w * x);
  R[2][2] = 1.0f - 2.0f * (x * x + y * y);

  float* o = &RT[k * 12];
  for (int i = 0; i < 3; ++i) {
    for (int j = 0; j < 3; ++j) o[i * 3 + j] = R[i][j];
    o[9 + i] = m2[i] - (R[i][0] * m1[0] + R[i][1] * m1[1] + R[i][2] * m1[2]);
  }
}

// ---------------------------------------------------------------------------
// Phase 3a: base streaming pass over ALL 3N elements, flat + b128 loads:
// Σ |(pc+sf) - (pc_t+sf_t)|. This carries most of the HBM traffic.
// ---------------------------------------------------------------------------
__global__ __launch_bounds__(256) void rigid_loss_base(
    const float* __restrict__ pc, const float* __restrict__ sf,
    const float* __restrict__ pc_t, const float* __restrict__ sf_t,
    float* __restrict__ part) {
  __shared__ float red[256];
  const int tid = threadIdx.x;
  float s = 0.0f;
  const int total4 = (3 * N_PTS) / 4;  // 3N divisible by 4
  const int stride = gridDim.x * blockDim.x;
  for (int j = blockIdx.x * blockDim.x + tid; j < total4; j += stride) {
    const v4f a  = *(const v4f*)(pc + 4 * j);
    const v4f b  = *(const v4f*)(sf + 4 * j);
    const v4f c  = *(const v4f*)(pc_t + 4 * j);
    const v4f d  = *(const v4f*)(sf_t + 4 * j);
#pragma unroll
    for (int e = 0; e < 4; ++e) s += fabsf((a[e] + b[e]) - (c[e] + d[e]));
  }
  red[tid] = s;
  __syncthreads();
  for (int o = 128; o > 0; o >>= 1) {
    if (tid < o) red[tid] += red[tid + o];
    __syncthreads();
  }
  if (tid == 0) part[blockIdx.x] = red[0];
}

// ---------------------------------------------------------------------------
// Phase 3b: correction over the K*M clustered points:
//   += |pred - (R*src + t)| - |pred - recon_t|
// ---------------------------------------------------------------------------
__global__ __launch_bounds__(256) void rigid_loss_fix(
    const float* __restrict__ pc, const float* __restrict__ sf,
    const float* __restrict__ pc_t, const float* __restrict__ sf_t,
    const int* __restrict__ cidx, const float* __restrict__ RT,
    float* __restrict__ part) {
  __shared__ float red[256];
  const int tid = threadIdx.x;
  float s = 0.0f;
  const int stride = gridDim.x * blockDim.x;
  for (int e = blockIdx.x * blockDim.x + tid; e < K_CL * M_PTS; e += stride) {
    const int k = e >> 13;  // e / M_PTS
    const int idx = cidx[e];
    const float* o = &RT[k * 12];
    const float sx = pc_t[idx];
    const float sy = pc_t[N_PTS + idx];
    const float sz = pc_t[2 * N_PTS + idx];
#pragma unroll
    for (int c = 0; c < 3; ++c) {
      const int off = c * N_PTS + idx;
      const float pred  = pc[off] + sf[off];
      const float rec   = pc_t[off] + sf_t[off];
      const float rigid = o[c * 3 + 0] * sx + o[c * 3 + 1] * sy +
                          o[c * 3 + 2] * sz + o[9 + c];
      s += fabsf(pred - rigid) - fabsf(pred - rec);
    }
  }
  red[tid] = s;
  __syncthreads();
  for (int o2 = 128; o2 > 0; o2 >>= 1) {
    if (tid < o2) red[tid] += red[tid + o2];
    __syncthreads();
  }
  if (tid == 0) part[blockIdx.x] = red[0];
}

// ---------------------------------------------------------------------------
// Phase 4: deterministic final reduction of all partials -> mean.
// ---------------------------------------------------------------------------
__global__ __launch_bounds__(256) void rigid_finalize(
    const float* __restrict__ part, float* __restrict__ out) {
  __shared__ float red[256];
  const int tid = threadIdx.x;
  float s = 0.0f;
  for (int i = tid; i < PART_A + PART_B; i += 256) s += part[i];
  red[tid] = s;
  __syncthreads();
  for (int o = 128; o > 0; o >>= 1) {
    if (tid < o) red[tid] += red[tid + o];
    __syncthreads();
  }
  if (tid == 0) out[0] = red[0] / (3.0f * (float)N_PTS);
}

extern "C" void kernel_launch(void* const* d_in, const int* in_sizes, int n_in,
                              void* d_out, int out_size, void* d_ws, size_t ws_size,
                              hipStream_t stream) {
  (void)in_sizes; (void)n_in; (void)out_size; (void)ws_size;
  const float* pc   = (const float*)d_in[0];
  const float* sf   = (const float*)d_in[1];
  const float* pc_t = (const float*)d_in[2];
  const float* sf_t = (const float*)d_in[3];
  const int*   cidx = (const int*)d_in[4];
  float* out = (float*)d_out;

  float* W    = (float*)d_ws;
  float* sums = W;                      // K*16 floats
  float* RT   = sums + K_CL * 16;       // K*12 floats (R row-major + t)
  float* part = RT + K_CL * 12;         // PART_A + PART_B floats

  rigid_phase1_sums<<<K_CL, 256, 0, stream>>>(pc_t, sf_t, cidx, sums);
  rigid_phase2_kabsch<<<1, 256, 0, stream>>>(sums, RT);
  rigid_loss_base<<<PART_A, 256, 0, stream>>>(pc, sf, pc_t, sf_t, part);
  rigid_loss_fix<<<PART_B, 256, 0, stream>>>(pc, sf, pc_t, sf_t, cidx, RT,
                                             part + PART_A);
  rigid_finalize<<<1, 256, 0, stream>>>(part, out);
}